// AdderLinear_89017492177334
// MI455X (gfx1250) — compile-verified
//
#include <hip/hip_runtime.h>
#include <cstdint>

// AdderLinear: out[n,o] = -|eta| * sum_i |x[n,i] - w[o,i]|
// N=2048, IN_F=1024, OUT_F=2048. VALU-bound (~8.6G VALU ops vs ~1.4us of HBM
// traffic), so: 2-op inner kernel (v_sub + v_add with |src| modifier), 4x4
// register tile per thread, 64x64 LDS tile per block, double-buffered
// CDNA5 async global->LDS loads tracked with ASYNCcnt.

#define N_ROWS 2048
#define IN_F   1024
#define OUT_F  2048
#define TILE   64
#define BK     16
#define LDS_W  28              // padded row (28-bank step: conflict-free w reads; 112B is 16B aligned)
#define NSTAGE (IN_F / BK)     // 64

__global__ __launch_bounds__(256) void AdderLinear_kernel(
    const float* __restrict__ x, const float* __restrict__ w,
    const float* __restrict__ eta, float* __restrict__ out)
{
    // double-buffered tiles, rows padded to LDS_W floats
    __shared__ __align__(16) float xs[2][TILE][LDS_W];
    __shared__ __align__(16) float ws[2][TILE][LDS_W];

    const int t    = threadIdx.x;
    const int row0 = blockIdx.y * TILE;   // over N
    const int col0 = blockIdx.x * TILE;   // over OUT_F

    // ---- loader mapping: one float4 of x and one of w per thread per stage
    const int lrow = t >> 2;    // 0..63  tile row
    const int lq   = t & 3;     // 0..3   quarter of the BK=16 floats

    const uint32_t xg0 = (uint32_t)(((row0 + lrow) * IN_F + lq * 4) * sizeof(float));
    const uint32_t wg0 = (uint32_t)(((col0 + lrow) * IN_F + lq * 4) * sizeof(float));
    const uint64_t xbase = (uint64_t)(uintptr_t)x;
    const uint64_t wbase = (uint64_t)(uintptr_t)w;

    // LDS byte offsets (low 32 bits of the flat address == LDS offset)
    uint32_t xl[2], wl[2];
    xl[0] = (uint32_t)(uintptr_t)&xs[0][lrow][lq * 4];
    xl[1] = (uint32_t)(uintptr_t)&xs[1][lrow][lq * 4];
    wl[0] = (uint32_t)(uintptr_t)&ws[0][lrow][lq * 4];
    wl[1] = (uint32_t)(uintptr_t)&ws[1][lrow][lq * 4];

    // ---- compute mapping: 16x16 thread grid, 4x4 outputs per thread
    const int tn = t & 15;
    const int tm = t >> 4;
    const int m0 = tm * 4;
    const int n0 = tn * 4;

    const float scale = -__builtin_fabsf(eta[0]);

    float acc[4][4];
#pragma unroll
    for (int i = 0; i < 4; ++i)
#pragma unroll
        for (int j = 0; j < 4; ++j) acc[i][j] = 0.0f;

    // ---- prologue: stage 0 async loads (ASYNCcnt += 2 per wave)
    asm volatile("global_load_async_to_lds_b128 %0, %1, %2"
                 :: "v"(xl[0]), "v"(xg0), "s"(xbase) : "memory");
    asm volatile("global_load_async_to_lds_b128 %0, %1, %2"
                 :: "v"(wl[0]), "v"(wg0), "s"(wbase) : "memory");

    int buf = 0;
    for (int s = 0; s < NSTAGE; ++s) {
        // issue stage s+1 into the other buffer, then wait for stage s
        if (s + 1 < NSTAGE) {
            const uint32_t koff = (uint32_t)((s + 1) * BK * sizeof(float));
            asm volatile("global_load_async_to_lds_b128 %0, %1, %2"
                         :: "v"(xl[buf ^ 1]), "v"(xg0 + koff), "s"(xbase) : "memory");
            asm volatile("global_load_async_to_lds_b128 %0, %1, %2"
                         :: "v"(wl[buf ^ 1]), "v"(wg0 + koff), "s"(wbase) : "memory");
            asm volatile("s_wait_asynccnt 2" ::: "memory");   // stage-s ops (oldest, in order) done
        } else {
            asm volatile("s_wait_asynccnt 0" ::: "memory");
        }
        __syncthreads();   // every wave's stage-s async writes are now in LDS

        // ---- 4x4x16 register-tile compute: acc += |x - w|
#pragma unroll
        for (int kk = 0; kk < BK; kk += 4) {
            float4 xv[4], wv[4];
#pragma unroll
            for (int i = 0; i < 4; ++i)
                xv[i] = *(const float4*)&xs[buf][m0 + i][kk];
#pragma unroll
            for (int j = 0; j < 4; ++j)
                wv[j] = *(const float4*)&ws[buf][n0 + j][kk];
#pragma unroll
            for (int i = 0; i < 4; ++i)
#pragma unroll
                for (int j = 0; j < 4; ++j) {
                    acc[i][j] += __builtin_fabsf(xv[i].x - wv[j].x);
                    acc[i][j] += __builtin_fabsf(xv[i].y - wv[j].y);
                    acc[i][j] += __builtin_fabsf(xv[i].z - wv[j].z);
                    acc[i][j] += __builtin_fabsf(xv[i].w - wv[j].w);
                }
        }

        __syncthreads();   // all reads of `buf` done before stage s+2 overwrites it
        buf ^= 1;
    }

    // ---- epilogue: out = -|eta| * dist
#pragma unroll
    for (int i = 0; i < 4; ++i) {
        float4 o;
        o.x = acc[i][0] * scale;
        o.y = acc[i][1] * scale;
        o.z = acc[i][2] * scale;
        o.w = acc[i][3] * scale;
        *(float4*)&out[(size_t)(row0 + m0 + i) * OUT_F + (col0 + n0)] = o;
    }
}

extern "C" void kernel_launch(void* const* d_in, const int* in_sizes, int n_in,
                              void* d_out, int out_size, void* d_ws, size_t ws_size,
                              hipStream_t stream) {
    const float* x   = (const float*)d_in[0];
    const float* w   = (const float*)d_in[1];
    const float* eta = (const float*)d_in[2];
    float* out = (float*)d_out;

    dim3 grid(OUT_F / TILE, N_ROWS / TILE);   // 32 x 32 workgroups
    AdderLinear_kernel<<<grid, 256, 0, stream>>>(x, w, eta, out);
}